// MaxKGINConv_51161650430039
// MI455X (gfx1250) — compile-verified
//
#include <hip/hip_runtime.h>

// GINConv (sum aggregator):
//   out[v] = (1+eps)*feat[v] + sum_{(u->v) in E} feat[u]
// N=100000 nodes, D=32 dims, E=1.6M edges, fp32.
//
// MI455X strategy:
//  - feat (12.8MB) + out (12.8MB) are L2-resident (192MB L2); the ~2x205MB of
//    gather/atomic traffic never leaves L2. HBM sees only ~38MB (~1.6us at
//    23.3 TB/s). The ceiling is L2 f32-atomic throughput, so the kernel is
//    built to issue exactly one coalesced 128B gather + one coalesced 128B of
//    device-scope hardware f32 atomics per edge and nothing else.
//  - WMMA deliberately NOT used: adjacency density is 1.6e-4; tiling this as
//    dense 16x16 matrix ops would inflate work >1000x with no traffic savings.
//  - wave32 mapping: lane == feature dim. Each wave handles 8 consecutive
//    edges; indices are wave-uniform (readfirstlane) so both index arrays are
//    fetched with one s_load_b256 each. global_prefetch_b8 streams the index
//    arrays ahead.

#define D_FEAT 32
#define WAVES_PER_BLOCK 8
#define EDGES_PER_WAVE 8
#define PREFETCH_EDGES 2048   // ~8KB ahead per int32 stream

__global__ void gin_init_kernel(const float* __restrict__ feat,
                                const float* __restrict__ eps,
                                float* __restrict__ out,
                                int n4 /* total elements / 4 */) {
    int i = blockIdx.x * blockDim.x + threadIdx.x;
    if (i >= n4) return;
    float s = 1.0f + eps[0];
    float4 f = reinterpret_cast<const float4*>(feat)[i];
    float4 o;
    o.x = s * f.x;
    o.y = s * f.y;
    o.z = s * f.z;
    o.w = s * f.w;
    reinterpret_cast<float4*>(out)[i] = o;
}

__device__ __forceinline__ void atomic_add_f32_dev(float* addr, float v) {
    // Non-returning global f32 atomic add, device scope (coherent across WGPs,
    // performed by the L2 atomic units). Inline asm guarantees the hardware
    // instruction (no CAS-loop fallback).
    asm volatile("global_atomic_add_f32 %0, %1, off scope:SCOPE_DEV"
                 :
                 : "v"(addr), "v"(v)
                 : "memory");
}

__global__ void gin_scatter_kernel(const float* __restrict__ feat,
                                   const int* __restrict__ edge_src,
                                   const int* __restrict__ edge_dst,
                                   float* __restrict__ out,
                                   int n_edges) {
    const int lane = threadIdx.x & 31;           // feature dim handled by this lane
    const int wave = threadIdx.x >> 5;           // wave id within block (0..7)

    // Wave-uniform base edge index -> SGPR; the 8 consecutive src (resp. dst)
    // indices of this wave are then fetched with a single s_load_b256.
    int ebase = __builtin_amdgcn_readfirstlane(
        (blockIdx.x * WAVES_PER_BLOCK + wave) * EDGES_PER_WAVE);

    if (ebase >= n_edges) return;

    // Stream the index arrays ahead into cache (global_prefetch_b8).
    if (ebase + PREFETCH_EDGES < n_edges) {
        __builtin_prefetch(&edge_src[ebase + PREFETCH_EDGES], 0, 1);
        __builtin_prefetch(&edge_dst[ebase + PREFETCH_EDGES], 0, 1);
    }

#pragma unroll
    for (int k = 0; k < EDGES_PER_WAVE; ++k) {
        int e = ebase + k;
        if (e < n_edges) {
            int s = edge_src[e];                          // scalar load (uniform)
            int d = edge_dst[e];                          // scalar load (uniform)
            float v = feat[(size_t)s * D_FEAT + lane];    // coalesced 128B/wave
            atomic_add_f32_dev(out + (size_t)d * D_FEAT + lane, v);
        }
    }
}

extern "C" void kernel_launch(void* const* d_in, const int* in_sizes, int n_in,
                              void* d_out, int out_size, void* d_ws, size_t ws_size,
                              hipStream_t stream) {
    const float* feat     = (const float*)d_in[0];  // [N, 32] fp32
    const float* eps      = (const float*)d_in[1];  // [1] fp32
    const int*   edge_src = (const int*)d_in[2];    // [E] int32
    const int*   edge_dst = (const int*)d_in[3];    // [E] int32
    float*       out      = (float*)d_out;          // [N, 32] fp32

    const int n_elems = in_sizes[0];        // N * 32
    const int n4      = n_elems / 4;
    const int n_edges = in_sizes[2];        // E

    // Phase 1: out = (1 + eps) * feat   (fully overwrites d_out -> no stale state)
    {
        const int threads = 256;
        const int blocks  = (n4 + threads - 1) / threads;
        gin_init_kernel<<<blocks, threads, 0, stream>>>(feat, eps, out, n4);
    }

    // Phase 2: scatter-add neighbor features with device-scope f32 atomics.
    {
        const int threads = WAVES_PER_BLOCK * 32;                      // 256
        const int edges_per_block = WAVES_PER_BLOCK * EDGES_PER_WAVE;  // 64
        const int blocks = (n_edges + edges_per_block - 1) / edges_per_block;
        gin_scatter_kernel<<<blocks, threads, 0, stream>>>(feat, edge_src, edge_dst,
                                                           out, n_edges);
    }
}